// pbDecoder_29068338659676
// MI455X (gfx1250) — compile-verified
//
#include <hip/hip_runtime.h>
#include <math.h>

typedef __attribute__((ext_vector_type(2))) float v2f;
typedef __attribute__((ext_vector_type(8))) float v8f;

#define NPTS   1920
#define GRID_N 33554432u   // 16 * 128^3

// ---------------------------------------------------------------------------
// Zero the two scatter grids (2 * GRID_N floats) in workspace.
// ---------------------------------------------------------------------------
__global__ void zero_kernel(float4* __restrict__ p, unsigned n4) {
  unsigned i = blockIdx.x * blockDim.x + threadIdx.x;
  if (i < n4) p[i] = make_float4(0.f, 0.f, 0.f, 0.f);
}

// ---------------------------------------------------------------------------
// InstanceNorm(affine=False) + ReLU over LDS buffer s viewed as (C, L),
// population variance, eps=1e-5.  256 threads, C in {16,32,64}.
// ---------------------------------------------------------------------------
__device__ inline void inorm_relu(float* s, int C, int L,
                                  float* red, float* red2,
                                  float* mean, float* inv, int tid) {
  int G = 256 / C;
  int c = tid / G;
  int r = tid - c * G;
  float sm = 0.f, sq = 0.f;
  for (int p = r; p < L; p += G) {
    float v = s[c * L + p];
    sm += v; sq += v * v;
  }
  red[tid] = sm; red2[tid] = sq;
  __syncthreads();
  if (r == 0) {
    float S = 0.f, Q = 0.f;
    for (int t = 0; t < G; ++t) { S += red[c * G + t]; Q += red2[c * G + t]; }
    float m = S / (float)L;
    float v = Q / (float)L - m * m;
    mean[c] = m;
    inv[c]  = rsqrtf(v + 1e-5f);
  }
  __syncthreads();
  int total = C * L;
  for (int i = tid; i < total; i += 256) {
    int cc = i / L;
    float y = (s[i] - mean[cc]) * inv[cc];
    s[i] = y > 0.f ? y : 0.f;
  }
  __syncthreads();
}

// ---------------------------------------------------------------------------
// Per-batch MLP (layer1 as f32 WMMA GEMM across 8 waves) + trilinear scatter
// of 1920 points into gridV (sigmoid vals) and gridO (ones).  1 block/batch.
// ---------------------------------------------------------------------------
__global__ __launch_bounds__(256)
void mlp_splat_kernel(const float* __restrict__ q,  const float* __restrict__ qval,
                      const float* __restrict__ w1, const float* __restrict__ b1,
                      const float* __restrict__ w2, const float* __restrict__ b2,
                      const float* __restrict__ w3, const float* __restrict__ b3,
                      const float* __restrict__ w4, const float* __restrict__ b4,
                      const float* __restrict__ w5, const float* __restrict__ b5,
                      float* __restrict__ gridV, float* __restrict__ gridO) {
  __shared__ float s_q16[4096];    // X (256 x 16), cols 8..15 zero-padded
  __shared__ float s_a[7680];      // layer1 out (1920) -> layer3 out (7680)
  __shared__ float s_b[3840];      // layer2 out
  __shared__ float red[256], red2[256];
  __shared__ float s_mean[64], s_inv[64];

  const int tid = threadIdx.x;
  const int b   = blockIdx.x;

  // Load X into a 16-wide zero-padded layout: s_q16[i*16 + l]
  for (int i = tid; i < 4096; i += 256) {
    int l = i & 15;
    s_q16[i] = (l < 8) ? q[b * 2048 + (i >> 4) * 8 + l] : 0.f;
  }
  __syncthreads();

  // ---- Layer 1: out(240x8) = w1(240x256) x X(256x8), WMMA f32 16x16x4 ----
  // 15 M-tiles distributed over 8 waves; EXEC is all-1 per wave in this loop.
  {
    const int lane  = tid & 31;
    const int wave  = tid >> 5;
    const int mrow  = lane & 15;          // A-row / B-col within tile
    const int khalf = (lane >> 4) << 1;   // K pair {0,1} or {2,3}
    for (int mt = wave; mt < 15; mt += 8) {
      v8f acc = {0.f, 0.f, 0.f, 0.f, 0.f, 0.f, 0.f, 0.f};
      const float* wrow = w1 + (mt * 16 + mrow) * 256;
      for (int k = 0; k < 256; k += 4) {
        v2f A, Bv;
        A.x  = wrow[k + khalf];
        A.y  = wrow[k + khalf + 1];
        Bv.x = s_q16[(k + khalf)     * 16 + mrow];   // unconditional: padded
        Bv.y = s_q16[(k + khalf + 1) * 16 + mrow];
        acc = __builtin_amdgcn_wmma_f32_16x16x4_f32(
                  false, A, false, Bv, (short)0, acc, false, false);
      }
      if (mrow < 8) {   // only N<8 columns are real
        const int l = mrow;
        #pragma unroll
        for (int r = 0; r < 8; ++r) {
          int o = mt * 16 + r + ((lane >> 4) << 3);   // C/D: M = r + 8*(lane/16)
          s_a[o * 8 + l] = acc[r] + b1[o];
        }
      }
    }
  }
  __syncthreads();

  // ---- IN(64,30) + ReLU ----
  inorm_relu(s_a, 64, 30, red, red2, s_mean, s_inv, tid);

  // ---- Layer 2: grouped conv 64->128 (scalar), out (64,2,30)=3840 ----
  for (int i = tid; i < 3840; i += 256) {
    int c = i / 60; int rem = i - c * 60; int j = rem / 30; int p = rem - j * 30;
    s_b[i] = s_a[c * 30 + p] * w2[c * 2 + j] + b2[c * 2 + j];
  }
  __syncthreads();
  inorm_relu(s_b, 32, 120, red, red2, s_mean, s_inv, tid);

  // ---- Layer 3: grouped conv 32->64, out (32,2,120)=7680 ----
  for (int i = tid; i < 7680; i += 256) {
    int c = i / 240; int rem = i - c * 240; int j = rem / 120; int p = rem - j * 120;
    s_a[i] = s_b[c * 120 + p] * w3[c * 2 + j] + b3[c * 2 + j];
  }
  __syncthreads();
  inorm_relu(s_a, 16, 480, red, red2, s_mean, s_inv, tid);

  // ---- Layer 4 IN stats computed analytically (output never materialized) --
  {
    int c = tid >> 4; int r = tid & 15;
    float sm = 0.f, sq = 0.f;
    for (int p = r; p < 480; p += 16) { float v = s_a[c * 480 + p]; sm += v; sq += v * v; }
    red[tid] = sm; red2[tid] = sq;
    __syncthreads();
    if (r == 0) {
      float S1 = 0.f, S2 = 0.f;
      for (int t = 0; t < 16; ++t) { S1 += red[c * 16 + t]; S2 += red2[c * 16 + t]; }
      float Ssum = 0.f, Sq = 0.f;
      for (int j = 0; j < 4; ++j) {
        float w = w4[c * 4 + j], bb = b4[c * 4 + j];
        Ssum += w * S1 + 480.f * bb;
        Sq   += w * w * S2 + 2.f * w * bb * S1 + 480.f * bb * bb;
      }
      float m = Ssum * (1.f / 1920.f);
      float v = Sq * (1.f / 1920.f) - m * m;
      s_mean[c] = m;
      s_inv[c]  = rsqrtf(v + 1e-5f);
    }
    __syncthreads();
  }

  // ---- Layer 4 apply + Layer 5 + tanh + trilinear scatter-add ----
  for (int n = tid; n < NPTS; n += 256) {
    int j = n / 480; int p = n - j * 480;
    float o0 = b5[0], o1 = b5[1], o2 = b5[2];
    #pragma unroll
    for (int i = 0; i < 16; ++i) {
      float t = s_a[i * 480 + p] * w4[i * 4 + j] + b4[i * 4 + j];
      t = (t - s_mean[i]) * s_inv[i];
      t = t > 0.f ? t : 0.f;
      o0 += w5[i]      * t;
      o1 += w5[16 + i] * t;
      o2 += w5[32 + i] * t;
    }
    float gx = tanhf(o0), gy = tanhf(o1), gz = tanhf(o2);
    float val = 1.f / (1.f + expf(-qval[n]));
    // px = ((g+1)*128 - 1)/2
    float px = (gx + 1.f) * 64.f - 0.5f;
    float py = (gy + 1.f) * 64.f - 0.5f;
    float pz = (gz + 1.f) * 64.f - 0.5f;
    float fx0 = floorf(px), fy0 = floorf(py), fz0 = floorf(pz);
    int x0 = (int)fx0, y0 = (int)fy0, z0 = (int)fz0;
    float fx = px - fx0, fy = py - fy0, fz = pz - fz0;
    unsigned base = (unsigned)b << 21;   // b * 128^3
    #pragma unroll
    for (int dz = 0; dz < 2; ++dz) {
      int zi = z0 + dz; if (zi < 0 || zi > 127) continue;
      float wz = dz ? fz : 1.f - fz;
      #pragma unroll
      for (int dy = 0; dy < 2; ++dy) {
        int yi = y0 + dy; if (yi < 0 || yi > 127) continue;
        float wy = dy ? fy : 1.f - fy;
        #pragma unroll
        for (int dx = 0; dx < 2; ++dx) {
          int xi = x0 + dx; if (xi < 0 || xi > 127) continue;
          float wv = wz * wy * (dx ? fx : 1.f - fx);
          unsigned flat = base + ((unsigned)zi << 14) + ((unsigned)yi << 7) + (unsigned)xi;
          atomicAdd(&gridV[flat], wv * val);
          atomicAdd(&gridO[flat], wv);
        }
      }
    }
  }
}

// ---------------------------------------------------------------------------
// One 3-tap average pass along an axis (zero boundary, count_include_pad).
// shift: 0 = x, 7 = y, 14 = z; axis length 128 so passes never cross batches.
// ---------------------------------------------------------------------------
__global__ __launch_bounds__(256)
void avg_pass(const float* __restrict__ src, float* __restrict__ dst, int shift) {
  unsigned idx = blockIdx.x * blockDim.x + threadIdx.x;
  if (idx >= GRID_N) return;
  unsigned stride = 1u << shift;
  unsigned c = (idx >> shift) & 127u;
  float v = src[idx];
  if (c != 0u)   v += src[idx - stride];
  if (c != 127u) v += src[idx + stride];
  dst[idx] = v * (1.f / 3.f);
}

// out = out / (ones_sm + 0.001)
__global__ __launch_bounds__(256)
void div_kernel(float* __restrict__ o, const float* __restrict__ den) {
  unsigned idx = blockIdx.x * blockDim.x + threadIdx.x;
  if (idx < GRID_N) o[idx] = o[idx] / (den[idx] + 0.001f);
}

// ---------------------------------------------------------------------------
extern "C" void kernel_launch(void* const* d_in, const int* in_sizes, int n_in,
                              void* d_out, int out_size, void* d_ws, size_t ws_size,
                              hipStream_t stream) {
  const float* q   = (const float*)d_in[0];
  const float* qv  = (const float*)d_in[1];
  const float* w1  = (const float*)d_in[2];
  const float* b1  = (const float*)d_in[3];
  const float* w2  = (const float*)d_in[4];
  const float* b2  = (const float*)d_in[5];
  const float* w3  = (const float*)d_in[6];
  const float* b3  = (const float*)d_in[7];
  const float* w4  = (const float*)d_in[8];
  const float* b4  = (const float*)d_in[9];
  const float* w5  = (const float*)d_in[10];
  const float* b5  = (const float*)d_in[11];

  float* W0 = (float*)d_ws;          // vals grid / ping-pong   (134 MB)
  float* W1 = W0 + GRID_N;           // ones grid               (134 MB)
  float* O  = (float*)d_out;         // final output, also temp (134 MB)

  // 1) zero both scatter grids
  unsigned n4 = (2u * GRID_N) / 4u;
  zero_kernel<<<n4 / 256u, 256, 0, stream>>>((float4*)d_ws, n4);

  // 2) MLP (WMMA) + trilinear scatter
  mlp_splat_kernel<<<16, 256, 0, stream>>>(q, qv, w1, b1, w2, b2, w3, b3,
                                           w4, b4, w5, b5, W0, W1);

  const unsigned nblk = GRID_N / 256u;
  const int shifts[3] = {14, 7, 0};   // z, y, x (reference axis order)

  // 3) smooth vals grid: 9 passes, ping-pong W0 <-> O, ends in O
  {
    float* src = W0; float* dst = O;
    for (int it = 0; it < 3; ++it)
      for (int a = 0; a < 3; ++a) {
        avg_pass<<<nblk, 256, 0, stream>>>(src, dst, shifts[a]);
        float* t = src; src = dst; dst = t;
      }
  }
  // 4) smooth ones grid: 9 passes, ping-pong W1 <-> W0, ends in W0
  {
    float* src = W1; float* dst = W0;
    for (int it = 0; it < 3; ++it)
      for (int a = 0; a < 3; ++a) {
        avg_pass<<<nblk, 256, 0, stream>>>(src, dst, shifts[a]);
        float* t = src; src = dst; dst = t;
      }
  }
  // 5) final division into d_out
  div_kernel<<<nblk, 256, 0, stream>>>(O, W0);
}